// TactIntentGNN_11845519802745
// MI455X (gfx1250) — compile-verified
//
#include <hip/hip_runtime.h>
#include <hip/hip_bf16.h>

#define Nn   16384
#define Ee   262144
#define Hh   4
#define Cc   128
#define HCc  512
#define Gg   8
#define INd  5
#define EDd  2
#define OUTd 256
#define NCLS 12

typedef float v2f __attribute__((ext_vector_type(2)));
typedef float v8f __attribute__((ext_vector_type(8)));
typedef int   v4i __attribute__((ext_vector_type(4)));

#if defined(__has_builtin)
#if __has_builtin(__builtin_amdgcn_global_load_async_to_lds_b128) && \
    __has_builtin(__builtin_amdgcn_s_wait_asynccnt)
#define HAVE_ASYNC_LDS 1
#endif
#endif

// -------------------- float atomic max (bit trick) --------------------
__device__ __forceinline__ void atomicMaxFloat(float* addr, float value) {
    if (value >= 0.0f)
        atomicMax((int*)addr, __float_as_int(value));
    else
        atomicMin((unsigned int*)addr, (unsigned int)__float_as_int(value));
}

// -------------------- async B-panel chunk loader --------------------
// Copies W[kc .. kc+128) x [col0 .. col0+64) into LDS with row stride 80 floats.
// 2048 16-byte segments, 8 per thread (256 threads).
__device__ __forceinline__ void load_b_chunk(const float* __restrict__ W, int Ncols,
                                             int col0, int kc, float* lbuf, int tid) {
#pragma unroll
    for (int i = 0; i < 8; ++i) {
        int seg = tid + (i << 8);
        int r = seg >> 4;              // 0..127
        int c4 = (seg & 15) << 2;      // 0,4,...,60
        const float* g = W + (long)(kc + r) * Ncols + col0 + c4;
        float* l = lbuf + r * 80 + c4;
#if defined(HAVE_ASYNC_LDS)
        __builtin_amdgcn_global_load_async_to_lds_b128(
            (__attribute__((address_space(1))) v4i*)g,
            (__attribute__((address_space(3))) v4i*)l, 0, 0);
#else
        *(float4*)l = *(const float4*)g;
#endif
    }
}

// -------------------- dense GEMM, fp32 WMMA 16x16x4, LDS-staged B --------------------
// Y[M,Ncols] = X[M,K] @ W[K,Ncols].
// Block = 8 waves: 128 rows x 64 cols; K consumed in double-buffered 128-chunks.
__global__ __launch_bounds__(256) void gemm_f32_wmma(
    const float* __restrict__ X, const float* __restrict__ W,
    float* __restrict__ Y, int M, int K, int Ncols) {
    __shared__ float Bs[2][128 * 80];
    const int colGroups = Ncols >> 6;
    const int blockRow = blockIdx.x / colGroups;
    const int colGroup = blockIdx.x - blockRow * colGroups;
    const int col0 = colGroup * 64;
    const int tid = threadIdx.x;
    const int wave = tid >> 5;
    const int lane = tid & 31;
    const int ln = lane & 15;
    const int hi = lane >> 4;                 // half-wave (K split)
    const int row0 = blockRow * 128 + wave * 16;

    const int nchunks = K >> 7;               // K/128
    load_b_chunk(W, Ncols, col0, 0, &Bs[0][0], tid);

    v8f c0 = {}, c1 = {}, c2 = {}, c3 = {};
    const float* xrow = X + (long)(row0 + ln) * K;

    for (int chunk = 0; chunk < nchunks; ++chunk) {
        const int cur = chunk & 1;
        if (chunk + 1 < nchunks) {
            load_b_chunk(W, Ncols, col0, (chunk + 1) << 7, &Bs[cur ^ 1][0], tid);
#if defined(HAVE_ASYNC_LDS)
            __builtin_amdgcn_s_wait_asynccnt(8);   // current chunk's 8 done
#endif
        } else {
#if defined(HAVE_ASYNC_LDS)
            __builtin_amdgcn_s_wait_asynccnt(0);
#endif
        }
        __syncthreads();

        const float* B0 = &Bs[cur][0];
        const float* xk = xrow + (chunk << 7) + hi * 2;
#pragma unroll 8
        for (int s = 0; s < 32; ++s) {
            v2f a = *(const v2f*)(xk + s * 4);              // K=ka, ka+1 (8B aligned)
            const float* r0 = B0 + (s * 4 + hi * 2) * 80;   // B row ka
            const float* r1 = r0 + 80;                      // B row ka+1
            v2f b0, b1, b2, b3;
            b0.x = r0[ln];      b0.y = r1[ln];
            b1.x = r0[ln + 16]; b1.y = r1[ln + 16];
            b2.x = r0[ln + 32]; b2.y = r1[ln + 32];
            b3.x = r0[ln + 48]; b3.y = r1[ln + 48];
            c0 = __builtin_amdgcn_wmma_f32_16x16x4_f32(false, a, false, b0, (short)0, c0, false, false);
            c1 = __builtin_amdgcn_wmma_f32_16x16x4_f32(false, a, false, b1, (short)0, c1, false, false);
            c2 = __builtin_amdgcn_wmma_f32_16x16x4_f32(false, a, false, b2, (short)0, c2, false, false);
            c3 = __builtin_amdgcn_wmma_f32_16x16x4_f32(false, a, false, b3, (short)0, c3, false, false);
        }
        __syncthreads();   // protect Bs[cur] before it is refilled
    }

    // C layout: VGPR v -> row = v + 8*hi, col = ln
#pragma unroll
    for (int v = 0; v < 8; ++v) {
        float* yr = Y + (long)(row0 + v + hi * 8) * Ncols + col0;
        yr[ln]      = c0[v];
        yr[ln + 16] = c1[v];
        yr[ln + 32] = c2[v];
        yr[ln + 48] = c3[v];
    }
}

// -------------------- layer-1 transforms (K=5, tiny) --------------------
__global__ __launch_bounds__(256) void lin_smallK(
    const float* __restrict__ x, const float* __restrict__ Wl,
    const float* __restrict__ Wr, float* __restrict__ xl, float* __restrict__ xr) {
    long i = blockIdx.x * 256L + threadIdx.x;
    if (i >= (long)Nn * HCc) return;
    int row = (int)(i >> 9);
    int col = (int)(i & (HCc - 1));
    float sl = 0.f, sr = 0.f;
#pragma unroll
    for (int k = 0; k < INd; ++k) {
        float xv = x[row * INd + k];
        sl += xv * Wl[k * HCc + col];
        sr += xv * Wr[k * HCc + col];
    }
    xl[i] = sl;
    xr[i] = sr;
}

// -------------------- per-layer init: m=-inf, denom=0, acc=bias --------------------
__global__ __launch_bounds__(256) void init_layer_kernel(
    float* __restrict__ mbuf, float* __restrict__ denom,
    float* __restrict__ acc, const float* __restrict__ bias) {
    long i = blockIdx.x * 256L + threadIdx.x;
    if (i < (long)Nn * Hh) { mbuf[i] = -__builtin_inff(); denom[i] = 0.f; }
    if (i >= (long)Nn * HCc) return;
    acc[i] = bias[i & (HCc - 1)];
}

// -------------------- edge logits + segment max (wave per edge) --------------------
__global__ __launch_bounds__(256) void edge_logits(
    const float* __restrict__ xl, const float* __restrict__ xr,
    const float* __restrict__ ea, const float* __restrict__ We,
    const float* __restrict__ att, const int* __restrict__ src,
    const int* __restrict__ dst, float* __restrict__ logit, float* __restrict__ mbuf) {
    int e = blockIdx.x * (blockDim.x >> 5) + (threadIdx.x >> 5);
    if (e >= Ee) return;
    int lane = threadIdx.x & 31;
    int s = src[e], d = dst[e];
    float e0 = ea[e * 2 + 0], e1 = ea[e * 2 + 1];
    int c0i = lane * 16;
    const float4* pl4 = (const float4*)(xl + (long)s * HCc + c0i);
    const float4* pr4 = (const float4*)(xr + (long)d * HCc + c0i);
    const float4* w04 = (const float4*)(We + c0i);
    const float4* w14 = (const float4*)(We + HCc + c0i);
    const float4* at4 = (const float4*)(att + c0i);
    float acc = 0.f;
#pragma unroll
    for (int j = 0; j < 4; ++j) {
        float4 pv = pl4[j], rv = pr4[j], w0 = w04[j], w1 = w14[j], av = at4[j];
        float v;
        v = pv.x + rv.x + e0 * w0.x + e1 * w1.x; v = v > 0.f ? v : 0.2f * v; acc += v * av.x;
        v = pv.y + rv.y + e0 * w0.y + e1 * w1.y; v = v > 0.f ? v : 0.2f * v; acc += v * av.y;
        v = pv.z + rv.z + e0 * w0.z + e1 * w1.z; v = v > 0.f ? v : 0.2f * v; acc += v * av.z;
        v = pv.w + rv.w + e0 * w0.w + e1 * w1.w; v = v > 0.f ? v : 0.2f * v; acc += v * av.w;
    }
    // reduce the 8 lanes belonging to one head (head = lane>>3)
    acc += __shfl_xor(acc, 1);
    acc += __shfl_xor(acc, 2);
    acc += __shfl_xor(acc, 4);
    if ((lane & 7) == 0) {
        int h = lane >> 3;
        logit[(long)e * Hh + h] = acc;
        atomicMaxFloat(&mbuf[(long)d * Hh + h], acc);
    }
}

// -------------------- exp(logit - max) + segment sum --------------------
__global__ __launch_bounds__(256) void edge_exp(
    const float* __restrict__ logit, const float* __restrict__ mbuf,
    const int* __restrict__ dst, float* __restrict__ aexp, float* __restrict__ denom) {
    long i = blockIdx.x * 256L + threadIdx.x;
    if (i >= (long)Ee * Hh) return;
    int e = (int)(i >> 2);
    int h = (int)(i & 3);
    int d = dst[e];
    float a = expf(logit[i] - mbuf[(long)d * Hh + h]);
    aexp[i] = a;
    atomicAdd(&denom[(long)d * Hh + h], a);
}

// -------------------- alpha-weighted scatter (wave per edge) --------------------
__global__ __launch_bounds__(256) void edge_scatter(
    const float* __restrict__ xl, const float* __restrict__ aexp,
    const float* __restrict__ denom, const int* __restrict__ src,
    const int* __restrict__ dst, float* __restrict__ out) {
    int e = blockIdx.x * (blockDim.x >> 5) + (threadIdx.x >> 5);
    if (e >= Ee) return;
    int lane = threadIdx.x & 31;
    int s = src[e], d = dst[e];
    int h = lane >> 3;
    float alpha = aexp[(long)e * Hh + h] / (denom[(long)d * Hh + h] + 1e-16f);
    int c0i = lane * 16;
    const float4* pl4 = (const float4*)(xl + (long)s * HCc + c0i);
    float* po = out + (long)d * HCc + c0i;
#pragma unroll
    for (int j = 0; j < 4; ++j) {
        float4 pv = pl4[j];
        atomicAdd(&po[j * 4 + 0], alpha * pv.x);
        atomicAdd(&po[j * 4 + 1], alpha * pv.y);
        atomicAdd(&po[j * 4 + 2], alpha * pv.z);
        atomicAdd(&po[j * 4 + 3], alpha * pv.w);
    }
}

// -------------------- zero helper --------------------
__global__ __launch_bounds__(256) void zero_f32(float* __restrict__ p, long n) {
    long i = blockIdx.x * 256L + threadIdx.x;
    if (i < n) p[i] = 0.f;
}

// -------------------- BN partial column sums (64 rows / block) --------------------
__global__ __launch_bounds__(256) void bn_partial(
    const float* __restrict__ x, float* __restrict__ sumb, float* __restrict__ sumsq) {
    int r0 = blockIdx.x * 64;
    int t = threadIdx.x;
    float s0 = 0.f, q0 = 0.f, s1 = 0.f, q1 = 0.f;
    for (int r = 0; r < 64; ++r) {
        const float* row = x + (long)(r0 + r) * HCc;
        float v0 = row[t], v1 = row[t + 256];
        s0 += v0; q0 += v0 * v0;
        s1 += v1; q1 += v1 * v1;
    }
    atomicAdd(&sumb[t], s0);       atomicAdd(&sumsq[t], q0);
    atomicAdd(&sumb[t + 256], s1); atomicAdd(&sumsq[t + 256], q1);
}

// -------------------- BN + ELU (+ residual) --------------------
__global__ __launch_bounds__(256) void bn_elu(
    const float* __restrict__ xin, const float* __restrict__ sumb,
    const float* __restrict__ sumsq, const float* __restrict__ g,
    const float* __restrict__ b, const float* __restrict__ resid,
    float* __restrict__ out) {
    long i = blockIdx.x * 256L + threadIdx.x;
    if (i >= (long)Nn * HCc) return;
    int col = (int)(i & (HCc - 1));
    const float inv_n = 1.0f / (float)Nn;
    float mu  = sumb[col] * inv_n;
    float var = sumsq[col] * inv_n - mu * mu;
    float y = g[col] * (xin[i] - mu) * rsqrtf(var + 1e-5f) + b[col];
    y = y > 0.f ? y : expm1f(y);                  // ELU(alpha=1)
    out[i] = resid ? (y + resid[i]) : y;
}

// -------------------- global mean pool (wave per node) --------------------
__global__ __launch_bounds__(256) void pool_kernel(
    const float* __restrict__ x3, const int* __restrict__ batch,
    float* __restrict__ pool, int* __restrict__ cnt) {
    int node = blockIdx.x * (blockDim.x >> 5) + (threadIdx.x >> 5);
    if (node >= Nn) return;
    int lane = threadIdx.x & 31;
    int gidx = batch[node];
    if (lane == 0) atomicAdd(&cnt[gidx], 1);
    const float* p = x3 + (long)node * HCc;
    for (int j = lane; j < HCc; j += 32)
        atomicAdd(&pool[gidx * HCc + j], p[j]);
}

// -------------------- head MLP (M=8, single block) --------------------
__global__ __launch_bounds__(256) void head_kernel(
    const float* __restrict__ pool, const int* __restrict__ cnt,
    const float* __restrict__ W1, const float* __restrict__ b1,
    const float* __restrict__ W2, const float* __restrict__ b2,
    const float* __restrict__ cW, const float* __restrict__ cb,
    float* __restrict__ out) {   // [0,96): logits ; [96,96+2048): emb
    __shared__ float xp[Gg][HCc];
    __shared__ float h1[Gg][OUTd];
    __shared__ float h2[Gg][OUTd];
    int t = threadIdx.x;
    for (int i = t; i < Gg * HCc; i += 256) {
        int gq = i >> 9;
        xp[gq][i & (HCc - 1)] = pool[i] / (float)cnt[gq];
    }
    __syncthreads();
    for (int gq = 0; gq < Gg; ++gq) {
        float s = b1[t];
        for (int k = 0; k < HCc; ++k) s += xp[gq][k] * W1[k * OUTd + t];
        h1[gq][t] = s > 0.f ? s : 0.f;            // ReLU
    }
    __syncthreads();
    for (int gq = 0; gq < Gg; ++gq) {
        float s = b2[t];
        for (int k = 0; k < OUTd; ++k) s += h1[gq][k] * W2[k * OUTd + t];
        h2[gq][t] = s;
        out[Gg * NCLS + gq * OUTd + t] = s;       // emb
    }
    __syncthreads();
    if (t < NCLS) {
        for (int gq = 0; gq < Gg; ++gq) {
            float s = cb[t];
            for (int k = 0; k < OUTd; ++k) s += h2[gq][k] * cW[k * NCLS + t];
            out[gq * NCLS + t] = s;               // logits
        }
    }
}

extern "C" void kernel_launch(void* const* d_in, const int* in_sizes, int n_in,
                              void* d_out, int out_size, void* d_ws, size_t ws_size,
                              hipStream_t stream) {
    const float* x     = (const float*)d_in[0];
    const float* ea    = (const float*)d_in[1];
    const int*   eidx  = (const int*)d_in[2];
    const int*   batch = (const int*)d_in[3];
    const int*   srcp  = eidx;
    const int*   dstp  = eidx + Ee;

    const float *Wl[3], *Wr[3], *We[3], *att[3], *bias[3], *bng[3], *bnb[3];
    for (int L = 0; L < 3; ++L) {
        int base = 4 + L * 7;
        Wl[L]   = (const float*)d_in[base + 0];
        Wr[L]   = (const float*)d_in[base + 1];
        We[L]   = (const float*)d_in[base + 2];
        att[L]  = (const float*)d_in[base + 3];
        bias[L] = (const float*)d_in[base + 4];
        bng[L]  = (const float*)d_in[base + 5];
        bnb[L]  = (const float*)d_in[base + 6];
    }
    const float* geW1 = (const float*)d_in[25];
    const float* geb1 = (const float*)d_in[26];
    const float* geW2 = (const float*)d_in[27];
    const float* geb2 = (const float*)d_in[28];
    const float* clsW = (const float*)d_in[29];
    const float* clsb = (const float*)d_in[30];

    // ---- workspace carve-up (floats) ----
    float* ws = (float*)d_ws;
    size_t o = 0;
    float* xl    = ws + o; o += (size_t)Nn * HCc;
    float* xr    = ws + o; o += (size_t)Nn * HCc;
    float* acc   = ws + o; o += (size_t)Nn * HCc;
    float* x1    = ws + o; o += (size_t)Nn * HCc;
    float* x2    = ws + o; o += (size_t)Nn * HCc;
    float* x3    = ws + o; o += (size_t)Nn * HCc;
    float* logit = ws + o; o += (size_t)Ee * Hh;
    float* aexp  = ws + o; o += (size_t)Ee * Hh;
    float* mbuf  = ws + o; o += (size_t)Nn * Hh;
    float* denom = ws + o; o += (size_t)Nn * Hh;
    float* sumb  = ws + o; o += HCc;          // contiguous with sumsq
    float* sumsq = ws + o; o += HCc;
    float* pool  = ws + o; o += (size_t)Gg * HCc;  // contiguous with cnt
    int*   cnt   = (int*)(ws + o); o += Gg;

    const int gemmBlocks = (Nn / 128) * (HCc / 64);   // 128x64 tile per block
    float* louts[3] = {x1, x2, x3};

    for (int L = 0; L < 3; ++L) {
        if (L == 0) {
            lin_smallK<<<(Nn * HCc) / 256, 256, 0, stream>>>(x, Wl[0], Wr[0], xl, xr);
        } else {
            const float* in = (L == 1) ? x1 : x2;
            gemm_f32_wmma<<<gemmBlocks, 256, 0, stream>>>(in, Wl[L], xl, Nn, HCc, HCc);
            gemm_f32_wmma<<<gemmBlocks, 256, 0, stream>>>(in, Wr[L], xr, Nn, HCc, HCc);
        }
        init_layer_kernel<<<(Nn * HCc) / 256, 256, 0, stream>>>(mbuf, denom, acc, bias[L]);
        edge_logits<<<Ee / 8, 256, 0, stream>>>(xl, xr, ea, We[L], att[L], srcp, dstp, logit, mbuf);
        edge_exp<<<(Ee * Hh) / 256, 256, 0, stream>>>(logit, mbuf, dstp, aexp, denom);
        edge_scatter<<<Ee / 8, 256, 0, stream>>>(xl, aexp, denom, srcp, dstp, acc);
        zero_f32<<<(2 * HCc + 255) / 256, 256, 0, stream>>>(sumb, 2 * HCc);
        bn_partial<<<Nn / 64, 256, 0, stream>>>(acc, sumb, sumsq);
        const float* resid = (L == 0) ? nullptr : ((L == 1) ? x1 : x2);
        bn_elu<<<(Nn * HCc) / 256, 256, 0, stream>>>(acc, sumb, sumsq, bng[L], bnb[L], resid, louts[L]);
    }

    zero_f32<<<(Gg * HCc + Gg + 255) / 256, 256, 0, stream>>>(pool, Gg * HCc + Gg);
    pool_kernel<<<Nn / 8, 256, 0, stream>>>(x3, batch, pool, cnt);
    head_kernel<<<1, 256, 0, stream>>>(pool, cnt, geW1, geb1, geW2, geb2, clsW, clsb, (float*)d_out);
}